// TransitionModel_22101901705279
// MI455X (gfx1250) — compile-verified
//
#include <hip/hip_runtime.h>
#include <math.h>

#define T_DIM 1024
#define N_DIM 512

#define BK 64                     // K-chunk staged in LDS per iteration
#define NCHUNK (N_DIM / BK)       // 8
#define LDS_PAD 4                 // pad floats/row: lane m -> bank 4m (conflict-free)
#define LDS_STRIDE (BK + LDS_PAD) // 68 floats = 272 B

typedef float v2f __attribute__((ext_vector_type(2)));
typedef float v8f __attribute__((ext_vector_type(8)));

// ---------------------------------------------------------------------------
// Kernel 1: c[k] = logsumexp over rows i of U[i,k]  (column LSE; coalesced)
// ---------------------------------------------------------------------------
__global__ void col_lse_kernel(const float* __restrict__ U, float* __restrict__ c) {
    int k = blockIdx.x * blockDim.x + threadIdx.x;
    if (k >= N_DIM) return;
    float m = -INFINITY;
    for (int i = 0; i < N_DIM; ++i) m = fmaxf(m, U[i * N_DIM + k]);
    float s = 0.0f;
    for (int i = 0; i < N_DIM; ++i) s += expf(U[i * N_DIM + k] - m);
    c[k] = m + logf(s);
}

// ---------------------------------------------------------------------------
// Kernel 2: mA[t] = max_k log_alpha[t,k];  Aexp[t,k] = exp(log_alpha[t,k]-mA[t])
// ---------------------------------------------------------------------------
__global__ void prep_a_kernel(const float* __restrict__ LA,
                              float* __restrict__ Aexp,
                              float* __restrict__ mA) {
    __shared__ float red[256];
    const int t = blockIdx.x;
    const int tid = threadIdx.x;
    const float* row = LA + t * N_DIM;
    float v0 = row[tid];
    float v1 = row[tid + 256];
    red[tid] = fmaxf(v0, v1);
    __syncthreads();
    for (int s = 128; s > 0; s >>= 1) {
        if (tid < s) red[tid] = fmaxf(red[tid], red[tid + s]);
        __syncthreads();
    }
    const float mr = red[0];
    Aexp[t * N_DIM + tid]       = expf(v0 - mr);
    Aexp[t * N_DIM + tid + 256] = expf(v1 - mr);
    if (tid == 0) mA[t] = mr;
}

// ---------------------------------------------------------------------------
// Kernel 3: log_T row = U[i,:] - c[:];  mB[i] = max of row; Bexp = exp(row-mB)
// ---------------------------------------------------------------------------
__global__ void prep_b_kernel(const float* __restrict__ U,
                              const float* __restrict__ c,
                              float* __restrict__ Bexp,
                              float* __restrict__ mB) {
    __shared__ float red[256];
    const int i = blockIdx.x;
    const int tid = threadIdx.x;
    const float* row = U + i * N_DIM;
    float v0 = row[tid]       - c[tid];
    float v1 = row[tid + 256] - c[tid + 256];
    red[tid] = fmaxf(v0, v1);
    __syncthreads();
    for (int s = 128; s > 0; s >>= 1) {
        if (tid < s) red[tid] = fmaxf(red[tid], red[tid + s]);
        __syncthreads();
    }
    const float mr = red[0];
    Bexp[i * N_DIM + tid]       = expf(v0 - mr);
    Bexp[i * N_DIM + tid + 256] = expf(v1 - mr);
    if (tid == 0) mB[i] = mr;
}

// ---------------------------------------------------------------------------
// Kernel 4: double-buffered async-LDS WMMA GEMM
//   S = Aexp(1024x512) * Bexp^T(512x512);  out = mA[t] + mB[i] + log(S)
//
// Block = 256 threads (8 wave32), macro-tile 64(t) x 64(i). K staged in
// chunks of 64 via GLOBAL_LOAD_ASYNC_TO_LDS_B128 into double-buffered LDS
// (padded stride 68 -> conflict-free ds_load_b64 fragment reads). Chunk n+1's
// async copies are issued before computing chunk n; s_wait_asynccnt 0 +
// workgroup barrier publish the LDS writes.
// Wave w: t-strip=(w&3)*16, i-half=(w>>2)*32 -> 2 tiles, A fragment reused.
// f32 WMMA lane layout (ISA 7.12.2): lane L: m=L%16, h=L/16;
//   A/B fragment = row[m] floats [ks+2h, ks+2h+1]; D vgpr r -> (M=r+8h, N=m).
// ---------------------------------------------------------------------------
__global__ void __launch_bounds__(256)
wmma_lse_gemm_kernel(const float* __restrict__ Aexp,
                     const float* __restrict__ Bexp,
                     const float* __restrict__ mA,
                     const float* __restrict__ mB,
                     float* __restrict__ out) {
    __shared__ __attribute__((aligned(16))) float lA[2][64 * LDS_STRIDE];
    __shared__ __attribute__((aligned(16))) float lB[2][64 * LDS_STRIDE];

    const int tid  = threadIdx.x;
    const int lane = tid & 31;
    const int wave = tid >> 5;
    const int t0 = (blockIdx.x >> 3) * 64;   // 16 macro-tiles along T
    const int i0 = (blockIdx.x & 7) * 64;    //  8 macro-tiles along N(i)
    const int m = lane & 15;
    const int h = lane >> 4;                 // 0 or 1
    const int ws = (wave & 3) * 16;          // t-strip within macro-tile
    const int wj = (wave >> 2) * 32;         // i-half within macro-tile

    // staging map: chunk c = tid + 256*r (r=0..3): row = c/16, col = (c%16)*4
    const int srow = tid >> 4;               // 0..15 base row
    const int scol = (tid & 15) << 2;        // 0,4,...,60

    // Issue 4+4 async 128-bit global->LDS copies for one K-chunk.
    // LDS addr operand = low 32 bits of the flat address (ISA 10.2:
    // address truncated to 32 bits).
    auto stage = [&](int kk0, int buf) {
#pragma unroll
        for (int r = 0; r < 4; ++r) {
            const int row = srow + 16 * r;
            unsigned la_off = (unsigned)(size_t)&lA[buf][row * LDS_STRIDE + scol];
            unsigned lb_off = (unsigned)(size_t)&lB[buf][row * LDS_STRIDE + scol];
            const float* ga = Aexp + (size_t)(t0 + row) * N_DIM + kk0 + scol;
            const float* gb = Bexp + (size_t)(i0 + row) * N_DIM + kk0 + scol;
            asm volatile("global_load_async_to_lds_b128 %0, %1, off"
                         :: "v"(la_off), "v"(ga) : "memory");
            asm volatile("global_load_async_to_lds_b128 %0, %1, off"
                         :: "v"(lb_off), "v"(gb) : "memory");
        }
    };

    v8f acc0 = {};
    v8f acc1 = {};

    stage(0, 0);
    asm volatile("s_wait_asynccnt 0x0" ::: "memory");
    __syncthreads();

    for (int chunk = 0; chunk < NCHUNK; ++chunk) {
        const int cur = chunk & 1;
        // overlap: issue next chunk's copies into the other buffer
        if (chunk + 1 < NCHUNK) stage((chunk + 1) * BK, cur ^ 1);

        // ---- 16 K-steps of V_WMMA_F32_16X16X4_F32, 2 tiles per wave ----
        const float* la  = &lA[cur][(ws + m) * LDS_STRIDE + 2 * h];
        const float* lb0 = &lB[cur][(wj + m) * LDS_STRIDE + 2 * h];
        const float* lb1 = &lB[cur][(wj + 16 + m) * LDS_STRIDE + 2 * h];
#pragma unroll
        for (int ks = 0; ks < BK; ks += 4) {
            v2f a  = *(const v2f*)(la + ks);   // 8B aligned: 2h+ks even
            v2f b0 = *(const v2f*)(lb0 + ks);
            v2f b1 = *(const v2f*)(lb1 + ks);
            acc0 = __builtin_amdgcn_wmma_f32_16x16x4_f32(
                false, a, false, b0, (short)0, acc0, false, false);
            acc1 = __builtin_amdgcn_wmma_f32_16x16x4_f32(
                false, a, false, b1, (short)0, acc1, false, false);
        }

        // my async copies done, then publish to all waves
        asm volatile("s_wait_asynccnt 0x0" ::: "memory");
        __syncthreads();
    }

    // ---- epilogue: out[t,i] = mA[t] + mB[i] + log(acc) ----
    const float lb_0 = mB[i0 + wj + m];
    const float lb_1 = mB[i0 + wj + 16 + m];
#pragma unroll
    for (int r = 0; r < 8; ++r) {
        const int trow = t0 + ws + r + 8 * h;
        const float la_r = mA[trow];
        out[trow * N_DIM + i0 + wj + m]      = la_r + lb_0 + logf(acc0[r]);
        out[trow * N_DIM + i0 + wj + 16 + m] = la_r + lb_1 + logf(acc1[r]);
    }
}

// ---------------------------------------------------------------------------
// Launch
// ---------------------------------------------------------------------------
extern "C" void kernel_launch(void* const* d_in, const int* in_sizes, int n_in,
                              void* d_out, int out_size, void* d_ws, size_t ws_size,
                              hipStream_t stream) {
    const float* log_alpha = (const float*)d_in[0];   // [T, N]
    const float* U         = (const float*)d_in[1];   // [N, N]
    float* out             = (float*)d_out;           // [T, N]

    // Workspace layout:
    char* ws = (char*)d_ws;
    float* Aexp = (float*)(ws);                                   // T*N*4 = 2 MB
    float* Bexp = (float*)(ws + (size_t)T_DIM * N_DIM * 4);       // N*N*4 = 1 MB
    float* mA   = (float*)(ws + (size_t)(T_DIM + N_DIM) * N_DIM * 4);          // T*4
    float* mB   = (float*)(ws + (size_t)(T_DIM + N_DIM) * N_DIM * 4 + 4096);   // N*4
    float* c    = (float*)(ws + (size_t)(T_DIM + N_DIM) * N_DIM * 4 + 8192);   // N*4

    // 1) column logsumexp of U
    col_lse_kernel<<<dim3(N_DIM / 256), dim3(256), 0, stream>>>(U, c);
    // 2) shift+exp rows of log_alpha
    prep_a_kernel<<<dim3(T_DIM), dim3(256), 0, stream>>>(log_alpha, Aexp, mA);
    // 3) shift+exp rows of log_T (depends on c; same-stream ordering)
    prep_b_kernel<<<dim3(N_DIM), dim3(256), 0, stream>>>(U, c, Bexp, mB);
    // 4) double-buffered async-LDS WMMA GEMM: (T/64)*(N/64) = 128 blocks
    wmma_lse_gemm_kernel<<<dim3(128), dim3(256), 0, stream>>>(Aexp, Bexp, mA, mB, out);
}